// src_DGCNN_7284264534342
// MI455X (gfx1250) — compile-verified
//
#include <hip/hip_runtime.h>
#include <hip/hip_bf16.h>

// ---------------------------------------------------------------------------
// DGCNN forward for MI455X (gfx1250), f32 end-to-end via V_WMMA_F32_16X16X4_F32
//   Phase A: KNN  — distance tiles as one WMMA each (augmented rank-4 trick)
//   Phase B: edge convs 4->8->8->16->64 + BN/ReLU + MLP-over-K, per-point wave
//   Phase C: 96x96 head GEMM + BN/ReLU + L2 normalize
// ---------------------------------------------------------------------------

typedef __attribute__((ext_vector_type(2))) float v2f;
typedef __attribute__((ext_vector_type(8))) float v8f;

#define NPTS  4096
#define NB    8
#define KNN_K 16
#define FEATC 96
#define EPS   1e-5f

__device__ __forceinline__ v8f wmma4(v2f a, v2f b, v8f c) {
  // D = A(16x4) * B(4x16) + C(16x16), all f32
  return __builtin_amdgcn_wmma_f32_16x16x4_f32(
      /*neg_a=*/false, a, /*neg_b=*/false, b,
      /*c_mod=*/(short)0, c, /*reuse_a=*/false, /*reuse_b=*/false);
}

// ===========================================================================
// Kernel A: KNN (top-16 nearest incl. self), one wave per 16 queries.
//   A rows = candidates: [x, y, |x|^2, 1]
//   B cols = queries   : [-2x, -2y, 1, |x|^2]
//   D[i][j] = squared distance(candidate i, query j)
// C layout: lane l holds query (l&15); rows v+8*(l>>4). Each lane keeps a
// sorted top-16 of its half of the candidates; lanes l and l+16 merge via LDS.
// Per-lane candidate indices are strictly ascending, so a strict '<' bubble
// is stable (lower index first on ties). Insert is branchless (v_cndmask
// chain) behind a single cheap exec-skip guard.
// ===========================================================================
__global__ __launch_bounds__(32) void knn_kernel(const float* __restrict__ x,
                                                 int* __restrict__ idxout) {
  const int b    = blockIdx.x >> 8;          // 256 query tiles per batch
  const int q0   = (blockIdx.x & 255) << 4;
  const int lane = threadIdx.x;
  const int lo   = lane & 15;
  const int hi   = lane >> 4;
  const float2* xb = (const float2*)(x + (size_t)b * NPTS * 2);

  // B operand (queries), built once
  float2 q = xb[q0 + lo];
  v2f bm;
  if (hi == 0) { bm.x = -2.0f * q.x; bm.y = -2.0f * q.y; }
  else         { bm.x = 1.0f;        bm.y = q.x * q.x + q.y * q.y; }

  float ld[16];
  int   li[16];
#pragma unroll
  for (int j = 0; j < 16; ++j) { ld[j] = 3.4e38f; li[j] = 0x7fffffff; }

  for (int c0 = 0; c0 < NPTS; c0 += 16) {
    float2 cpt = xb[c0 + lo];
    v2f am;
    if (hi == 0) { am.x = cpt.x; am.y = cpt.y; }
    else         { am.x = cpt.x * cpt.x + cpt.y * cpt.y; am.y = 1.0f; }
    v8f acc = {};
    acc = wmma4(am, bm, acc);

#pragma unroll
    for (int v = 0; v < 8; ++v) {
      float d  = acc[v];
      int   id = c0 + v + 8 * hi;
      if (d < ld[15]) {                        // exec-skip guard only
        float cd = d; int ci = id;
#pragma unroll
        for (int j = 0; j < 16; ++j) {         // branchless stable bubble
          const bool  less = cd < ld[j];
          const float tf = ld[j];
          const int   ti = li[j];
          ld[j] = less ? cd : tf;
          li[j] = less ? ci : ti;
          cd    = less ? tf : cd;
          ci    = less ? ti : ci;
        }
      }
    }
  }

  __shared__ float sd[32][16];
  __shared__ int   si[32][16];
#pragma unroll
  for (int j = 0; j < 16; ++j) { sd[lane][j] = ld[j]; si[lane][j] = li[j]; }
  __syncthreads();

  if (lane < 16) {                            // merge the two half-lists
    int pa = 0, pb = 0;
    int* op = idxout + ((size_t)b * NPTS + q0 + lane) * KNN_K;
    for (int j = 0; j < 16; ++j) {
      float da = sd[lane][pa],  db = sd[lane + 16][pb];
      int   ia = si[lane][pa],  ib = si[lane + 16][pb];
      bool ta = (da < db) || (da == db && ia < ib);
      op[j] = ta ? ia : ib;
      if (ta) ++pa; else ++pb;
    }
  }
}

// ===========================================================================
// Kernel B: per-point edge convs. One wave per point, 8 waves / block.
// A matrices: M = 16 neighbors, K = input channels (4 per WMMA step).
// After each WMMA: BN+ReLU per lane-channel, MLP-over-K reduce (shfl_xor 16),
// then a 16x16 lane transpose through padded LDS to form the next A.
// ===========================================================================
__device__ __forceinline__ void layer_post(const v8f& d, const float* __restrict__ bn,
                                           int C, const float* __restrict__ mlpw_row,
                                           float mlpb, int lo, int hi,
                                           float* __restrict__ fout, float (*T)[17]) {
  float s = 0.0f, o = 0.0f;
  if (lo < C) {
    float g = bn[lo], be = bn[C + lo], mu = bn[2 * C + lo], va = bn[3 * C + lo];
    s = g * rsqrtf(va + EPS);
    o = be - mu * s;
  }
  float partial = 0.0f;
  float h[8];
#pragma unroll
  for (int v = 0; v < 8; ++v) {
    h[v] = fmaxf(d[v] * s + o, 0.0f);
    partial += h[v] * mlpw_row[v + 8 * hi];
  }
  float tot = partial + __shfl_xor(partial, 16, 32);
  if (hi == 0 && lo < C) fout[lo] = tot + mlpb;
#pragma unroll
  for (int v = 0; v < 8; ++v) T[v + 8 * hi][lo] = h[v];   // T[neighbor][channel]
}

__global__ __launch_bounds__(256) void edge_kernel(
    const float* __restrict__ x, const int* __restrict__ knn,
    const float* __restrict__ w1, const float* __restrict__ w2,
    const float* __restrict__ w3, const float* __restrict__ w4,
    const float* __restrict__ mlp_w, const float* __restrict__ mlp_b,
    const float* __restrict__ bn1, const float* __restrict__ bn2,
    const float* __restrict__ bn3, const float* __restrict__ bn4,
    float* __restrict__ feats) {
  const int wave = threadIdx.x >> 5;
  const int lane = threadIdx.x & 31;
  const int lo = lane & 15, hi = lane >> 4;
  const int p = blockIdx.x * 8 + wave;        // global point
  const int b = p >> 12, n = p & (NPTS - 1);
  const float* xb = x + (size_t)b * NPTS * 2;
  float* fp = feats + (size_t)p * FEATC;

  __shared__ float tb[8][16][17];
  float (*T)[17] = tb[wave];

  // ---- layer 1 input: [nbr.x, nbr.y, ctr.x, ctr.y] per neighbor row ----
  v2f a0;
  if (hi == 0) {
    int nb = knn[(size_t)p * KNN_K + lo];
    a0.x = xb[nb * 2 + 0]; a0.y = xb[nb * 2 + 1];
  } else {
    a0.x = xb[n * 2 + 0];  a0.y = xb[n * 2 + 1];
  }
  v2f b1 = {0.0f, 0.0f};                       // w1: [8,4]
  if (lo < 8) { b1.x = w1[lo * 4 + 2 * hi + 0]; b1.y = w1[lo * 4 + 2 * hi + 1]; }
  v8f d = {};
  d = wmma4(a0, b1, d);
  layer_post(d, bn1, 8, mlp_w + 0, mlp_b[0], lo, hi, fp + 0, T);
  __syncthreads();

  // ---- layer 2: 8 -> 8 ----
  v2f a2a, a2b;
  a2a.x = T[lo][2 * hi + 0]; a2a.y = T[lo][2 * hi + 1];
  a2b.x = T[lo][4 + 2 * hi + 0]; a2b.y = T[lo][4 + 2 * hi + 1];
  __syncthreads();
  v2f b2a = {0.0f, 0.0f}, b2b = {0.0f, 0.0f};  // w2: [8,8]
  if (lo < 8) {
    b2a.x = w2[lo * 8 + 2 * hi + 0];     b2a.y = w2[lo * 8 + 2 * hi + 1];
    b2b.x = w2[lo * 8 + 4 + 2 * hi + 0]; b2b.y = w2[lo * 8 + 4 + 2 * hi + 1];
  }
  d = v8f{};
  d = wmma4(a2a, b2a, d);
  d = wmma4(a2b, b2b, d);
  layer_post(d, bn2, 8, mlp_w + 16, mlp_b[1], lo, hi, fp + 8, T);
  __syncthreads();

  // ---- layer 3: 8 -> 16 ----
  v2f a3a, a3b;
  a3a.x = T[lo][2 * hi + 0]; a3a.y = T[lo][2 * hi + 1];
  a3b.x = T[lo][4 + 2 * hi + 0]; a3b.y = T[lo][4 + 2 * hi + 1];
  __syncthreads();
  v2f b3a, b3b;                                // w3: [16,8]
  b3a.x = w3[lo * 8 + 2 * hi + 0];     b3a.y = w3[lo * 8 + 2 * hi + 1];
  b3b.x = w3[lo * 8 + 4 + 2 * hi + 0]; b3b.y = w3[lo * 8 + 4 + 2 * hi + 1];
  d = v8f{};
  d = wmma4(a3a, b3a, d);
  d = wmma4(a3b, b3b, d);
  layer_post(d, bn3, 16, mlp_w + 32, mlp_b[2], lo, hi, fp + 16, T);
  __syncthreads();

  // ---- layer 4: 16 -> 64 (4 K-chunks x 4 output tiles) ----
  v2f a4[4];
#pragma unroll
  for (int kc = 0; kc < 4; ++kc) {
    a4[kc].x = T[lo][4 * kc + 2 * hi + 0];
    a4[kc].y = T[lo][4 * kc + 2 * hi + 1];
  }
#pragma unroll
  for (int t = 0; t < 4; ++t) {
    const int c = t * 16 + lo;
    v8f acc = {};
#pragma unroll
    for (int kc = 0; kc < 4; ++kc) {
      v2f bw;                                  // w4: [64,16]
      bw.x = w4[c * 16 + 4 * kc + 2 * hi + 0];
      bw.y = w4[c * 16 + 4 * kc + 2 * hi + 1];
      acc = wmma4(a4[kc], bw, acc);
    }
    float g = bn4[c], be = bn4[64 + c], mu = bn4[128 + c], va = bn4[192 + c];
    float s = g * rsqrtf(va + EPS), o = be - mu * s;
    float partial = 0.0f;
#pragma unroll
    for (int v = 0; v < 8; ++v) {
      float hv = fmaxf(acc[v] * s + o, 0.0f);
      partial += hv * mlp_w[3 * 16 + v + 8 * hi];
    }
    float tot = partial + __shfl_xor(partial, 16, 32);
    if (hi == 0) fp[32 + c] = tot + mlp_b[3];
  }
}

// ===========================================================================
// Kernel C: head GEMM [32768,96] x w5[96,96]^T + BN + ReLU + L2 normalize.
// One wave per 16 points: A (24 K-chunks) kept in 48 VGPRs, 6 col tiles.
// ===========================================================================
__global__ __launch_bounds__(256) void head_kernel(const float* __restrict__ feats,
                                                   const float* __restrict__ w5,
                                                   const float* __restrict__ bn5,
                                                   float* __restrict__ out) {
  const int wave = threadIdx.x >> 5;
  const int lane = threadIdx.x & 31;
  const int lo = lane & 15, hi = lane >> 4;
  const int r0 = (blockIdx.x * 8 + wave) * 16;

  v2f A[24];
  const float* fr = feats + (size_t)(r0 + lo) * FEATC + 2 * hi;
#pragma unroll
  for (int kc = 0; kc < 24; ++kc) { A[kc].x = fr[kc * 4 + 0]; A[kc].y = fr[kc * 4 + 1]; }

  float r[6][8];
#pragma unroll
  for (int t = 0; t < 6; ++t) {
    const int c = t * 16 + lo;
    const float* wr = w5 + (size_t)c * FEATC + 2 * hi;   // w5: [96,96]
    v8f acc = {};
#pragma unroll
    for (int kc = 0; kc < 24; ++kc) {
      v2f bw; bw.x = wr[kc * 4 + 0]; bw.y = wr[kc * 4 + 1];
      acc = wmma4(A[kc], bw, acc);
    }
    float g = bn5[c], be = bn5[96 + c], mu = bn5[192 + c], va = bn5[288 + c];
    float s = g * rsqrtf(va + EPS), o = be - mu * s;
#pragma unroll
    for (int v = 0; v < 8; ++v) r[t][v] = fmaxf(acc[v] * s + o, 0.0f);
  }

#pragma unroll
  for (int v = 0; v < 8; ++v) {
    float ss = 0.0f;
#pragma unroll
    for (int t = 0; t < 6; ++t) ss += r[t][v] * r[t][v];
    ss += __shfl_xor(ss, 1, 32);               // 16-lane butterfly (rows stay
    ss += __shfl_xor(ss, 2, 32);               //  within their half-wave)
    ss += __shfl_xor(ss, 4, 32);
    ss += __shfl_xor(ss, 8, 32);
    float inv = rsqrtf(ss);
    const int m = r0 + v + 8 * hi;
#pragma unroll
    for (int t = 0; t < 6; ++t) out[(size_t)m * FEATC + t * 16 + lo] = r[t][v] * inv;
  }
}

// ===========================================================================
extern "C" void kernel_launch(void* const* d_in, const int* in_sizes, int n_in,
                              void* d_out, int out_size, void* d_ws, size_t ws_size,
                              hipStream_t stream) {
  const float* x     = (const float*)d_in[0];
  const float* w1    = (const float*)d_in[1];
  const float* w2    = (const float*)d_in[2];
  const float* w3    = (const float*)d_in[3];
  const float* w4    = (const float*)d_in[4];
  const float* w5    = (const float*)d_in[5];
  const float* mlp_w = (const float*)d_in[6];
  const float* mlp_b = (const float*)d_in[7];
  const float* bn1   = (const float*)d_in[8];
  const float* bn2   = (const float*)d_in[9];
  const float* bn3   = (const float*)d_in[10];
  const float* bn4   = (const float*)d_in[11];
  const float* bn5   = (const float*)d_in[12];
  float* out = (float*)d_out;

  int*   knn   = (int*)d_ws;                                        // 2 MB
  float* feats = (float*)((char*)d_ws + (size_t)NB * NPTS * KNN_K * sizeof(int)); // 12 MB

  // A: 8 batches x 256 query tiles, one wave each
  knn_kernel<<<NB * (NPTS / 16), 32, 0, stream>>>(x, knn);
  // B: one wave per point, 8 waves per block
  edge_kernel<<<(NB * NPTS) / 8, 256, 0, stream>>>(x, knn, w1, w2, w3, w4,
                                                   mlp_w, mlp_b, bn1, bn2, bn3, bn4,
                                                   feats);
  // C: one wave per 16 points, 8 waves per block
  head_kernel<<<(NB * NPTS) / (16 * 8), 256, 0, stream>>>(feats, w5, bn5, out);
}